// DistanceTransformMap_49014166782432
// MI455X (gfx1250) — compile-verified
//
#include <hip/hip_runtime.h>
#include <cstdint>

// Problem constants (B,C,H,W = 4,1,384,384; C==1 so layout is [B*H, W] for rows)
#define Wn 384
#define Hn 384
#define Bn 4
#define TPB 384          // 12 wave32 waves per block
#define TROWS 16         // lines (rows or cols) per block
#define LARGE_F 589824.0f   // 2*(H*H+W*W)
#define BIN_THR 0.00196078431f  // 0.5/255: fg = (mask*255) > 0.5

// ---- TDM availability --------------------------------------------------
#if defined(__has_builtin)
#  if __has_builtin(__builtin_amdgcn_tensor_load_to_lds)
#    define HAVE_TDM 1
#  endif
#endif
#ifndef HAVE_TDM
#  define HAVE_TDM 0
#endif

typedef unsigned int u32x4 __attribute__((ext_vector_type(4)));
typedef int          i32x8 __attribute__((ext_vector_type(8)));
typedef int          i32x4 __attribute__((ext_vector_type(4)));

// Load a 2-D tile (tile0 x tile1 elements, f32) from global into contiguous LDS.
// gsrc points at the tile start; row_stride_elems is the dim-0 stride in elements.
// Uses the Tensor Data Mover when available (one wave issues, all waves sync).
__device__ __forceinline__ void load_tile_to_lds(float* lds, const float* gsrc,
                                                 uint32_t tile0, uint32_t tile1,
                                                 uint32_t row_stride_elems)
{
#if HAVE_TDM
    if (threadIdx.x < 32u) {   // one wave issues the DMA (EXEC ignored by TDM)
        const uint64_t ga  = (uint64_t)(uintptr_t)gsrc;
        const uint32_t lba = (uint32_t)(uintptr_t)lds;   // LDS aperture: low 32 bits = LDS byte addr

        // D# group 0 (ISA 8.3): count=1 | lds_addr | global_addr[56:0] | type=2
        u32x4 g0 = { 1u,
                     lba,
                     (uint32_t)ga,
                     (uint32_t)((ga >> 32) & 0x01FFFFFFu) | (2u << 30) };

        // D# group 1 (ISA 8.4): data_size=2 (4B), tensor_dim0/1 = tile extent,
        // tile_dim0/1, tensor_dim0_stride = row stride (elements). No pad/iterate/mask.
        const uint32_t td0 = tile0, td1 = tile1;
        const uint64_t s0  = (uint64_t)row_stride_elems;
        i32x8 g1 = { (int)(2u << 16),
                     (int)((td0 & 0xFFFFu) << 16),
                     (int)((td0 >> 16) | ((td1 & 0xFFFFu) << 16)),
                     (int)((td1 >> 16) | ((tile0 & 0xFFFFu) << 16)),
                     (int)(tile1 & 0xFFFFu),                 // tile_dim1 | tile_dim2=0
                     (int)(uint32_t)(s0 & 0xFFFFFFFFu),      // dim0 stride [31:0]
                     (int)(uint32_t)((s0 >> 32) & 0xFFFFu),  // dim0 stride [47:32] | dim1 stride lo = 0
                     0 };

        i32x4 z4 = { 0, 0, 0, 0 };
#if defined(__clang_major__) && (__clang_major__ >= 23)
        i32x8 z8 = { 0, 0, 0, 0, 0, 0, 0, 0 };
        __builtin_amdgcn_tensor_load_to_lds(g0, g1, z4, z4, z8, 0);
#else
        __builtin_amdgcn_tensor_load_to_lds(g0, g1, z4, z4, 0);
#endif
        __builtin_amdgcn_s_wait_tensorcnt(0);
    }
    __syncthreads();
#else
    for (uint32_t idx = threadIdx.x; idx < tile0 * tile1; idx += TPB) {
        uint32_t xx = idx % tile0, yy = idx / tile0;
        lds[idx] = gsrc[(size_t)yy * row_stride_elems + xx];
    }
    __syncthreads();
#endif
}

// Pass 1: per-row squared EDT along W.  out[r][i] = min_j f[r][j] + (i-j)^2,
// f = fg ? LARGE : 0.  One block = 16 consecutive rows of the [B*H, W] image.
__global__ void __launch_bounds__(TPB)
edt_rows_kernel(const float* __restrict__ mask, float* __restrict__ out)
{
    __shared__ float f[TROWS * Wn];          // 24 KB
    const int t    = threadIdx.x;
    const int row0 = blockIdx.x * TROWS;     // flat row in [0, B*H)

    load_tile_to_lds(f, mask + (size_t)row0 * Wn, Wn, TROWS, Wn);

    // Binarize in place: fg -> LARGE, bg -> 0 (hoists the compare out of the hot loop)
    for (int idx = t; idx < TROWS * Wn; idx += TPB)
        f[idx] = (f[idx] > BIN_THR) ? LARGE_F : 0.0f;
    __syncthreads();

    const int r  = t / 24;                   // 0..15 : row within tile
    const int i0 = t % 24;                   // lanes 0..23 share r -> LDS broadcast
    const float* fr = &f[r * Wn];

    for (int k = 0; k < Wn / 24; ++k) {
        const int   i  = i0 + 24 * k;
        const float fi = (float)i;
        float acc = 3.0e38f;
#pragma unroll 8
        for (int j = 0; j < Wn; ++j) {
            const float d = fi - (float)j;
            acc = fminf(acc, __builtin_fmaf(d, d, fr[j]));
        }
        out[(size_t)(row0 + r) * Wn + i] = acc;   // squared distance, no sqrt yet
    }
}

// Pass 2: per-column min-plus along H, then sqrt.  In place on the pass-1 result:
// each block owns a disjoint 16-column tile of one image, staged to LDS via TDM
// (strided gather: tile0=16 cols, tile1=H rows, dim0 stride = W).
__global__ void __launch_bounds__(TPB)
edt_cols_kernel(float* __restrict__ data)
{
    __shared__ float g[Hn * 16];             // 24 KB, g[y*16 + x]
    const int t  = threadIdx.x;
    const int b  = blockIdx.x / (Wn / 16);
    const int x0 = (blockIdx.x % (Wn / 16)) * 16;
    float* base  = data + (size_t)b * Hn * Wn + x0;

    load_tile_to_lds(g, base, 16, Hn, Wn);

    const int x  = t / 24;                   // 0..15 : column within tile
    const int i0 = t % 24;                   // lanes 0..23 share x -> LDS broadcast

    for (int k = 0; k < Hn / 24; ++k) {
        const int   i  = i0 + 24 * k;
        const float fi = (float)i;
        float acc = 3.0e38f;
#pragma unroll 8
        for (int y = 0; y < Hn; ++y) {
            const float d = fi - (float)y;
            acc = fminf(acc, __builtin_fmaf(d, d, g[y * 16 + x]));
        }
        base[(size_t)i * Wn + x] = sqrtf(acc);
    }
}

extern "C" void kernel_launch(void* const* d_in, const int* in_sizes, int n_in,
                              void* d_out, int out_size, void* d_ws, size_t ws_size,
                              hipStream_t stream)
{
    (void)in_sizes; (void)n_in; (void)d_ws; (void)ws_size; (void)out_size;
    const float* mask = (const float*)d_in[0];
    float* out        = (float*)d_out;

    const int rowBlocks = (Bn * Hn) / TROWS;   // 96
    const int colBlocks = Bn * (Wn / 16);      // 96

    edt_rows_kernel<<<rowBlocks, TPB, 0, stream>>>(mask, out);
    edt_cols_kernel<<<colBlocks, TPB, 0, stream>>>(out);  // in place: squared -> sqrt
}